// Bidirectional_Adaptive_Region_Selection_22376779612918
// MI455X (gfx1250) — compile-verified
//
#include <hip/hip_runtime.h>

#define NUM_CLASSES 19
#define FEAT_CH     512
#define C_PAD       32                 // two 16-class WMMA tiles
#define COL_TILES   (FEAT_CH / 16)     // 32
#define WPB         8                  // waves per block (256 threads, wave32)
#define CHUNK       4096               // pixels per wave-task

typedef float v2f __attribute__((ext_vector_type(2)));
typedef float v8f __attribute__((ext_vector_type(8)));

// ---------------------------------------------------------------- zero scratch
__global__ void zero_ws_kernel(float* __restrict__ ws, int n) {
    int i = blockIdx.x * blockDim.x + threadIdx.x;
    if (i < n) ws[i] = 0.0f;
}

// ---------------------------------------------------------------- label counts
__global__ void count_kernel(const int* __restrict__ labels, int n,
                             const int* __restrict__ ignore_p,
                             float* __restrict__ counts /*C_PAD*/) {
    __shared__ unsigned int hist[C_PAD];
    if (threadIdx.x < C_PAD) hist[threadIdx.x] = 0u;
    __syncthreads();
    const int ignore = *ignore_p;
    const int stride = gridDim.x * blockDim.x;
    for (int i = blockIdx.x * blockDim.x + threadIdx.x; i < n; i += stride) {
        int lab = labels[i];
        if (lab != ignore && lab >= 0 && lab < C_PAD)
            atomicAdd(&hist[lab], 1u);
    }
    __syncthreads();
    if (threadIdx.x < C_PAD) {
        unsigned int h = hist[threadIdx.x];
        if (h) atomicAdd(&counts[threadIdx.x], (float)h);
    }
}

// ------------------------------------------------- WMMA one-hot GEMM segment sum
// sums(C_PAD x 512) += OneHot^T(C_PAD x N) * F(N x 512), f32 exact via
// V_WMMA_F32_16X16X4_F32. One wave: one 16-col tile, both class tiles, CHUNK pixels.
__global__ __launch_bounds__(256)
void segsum_wmma_kernel(const float* __restrict__ feat,
                        const int*   __restrict__ labels,
                        const int*   __restrict__ ignore_p,
                        float*       __restrict__ sums /*C_PAD x FEAT_CH*/) {
    const int wave = threadIdx.x >> 5;
    const int lane = threadIdx.x & 31;
    const int l16  = lane & 15;
    const int hi   = lane >> 4;                 // 0: pixels {0,1}; 1: pixels {2,3}
    const int colb = blockIdx.y * 16;           // feature column tile base
    const int ignore = *ignore_p;

    const long p0 = (long)(blockIdx.x * WPB + wave) * CHUNK;

    // per-lane streaming pointers (advance by 4 pixels per iteration)
    const float* fptr = feat + (p0 + 2 * hi) * (long)FEAT_CH + colb + l16;
    const int*   lptr = labels + p0 + 2 * hi;

    const int c0 = l16;          // class in tile 0
    const int c1 = 16 + l16;     // class in tile 1
    const float w0 = (c0 != ignore) ? 1.0f : 0.0f;
    const float w1 = (c1 != ignore) ? 1.0f : 0.0f;

    v8f acc0 = {};
    v8f acc1 = {};

#pragma unroll 4
    for (int s = 0; s < CHUNK; s += 4) {
        const int labA = lptr[0];
        const int labB = lptr[1];
        v2f b;
        b.x = fptr[0];
        b.y = fptr[FEAT_CH];

        v2f a0, a1;
        a0.x = (labA == c0) ? w0 : 0.0f;
        a0.y = (labB == c0) ? w0 : 0.0f;
        a1.x = (labA == c1) ? w1 : 0.0f;
        a1.y = (labB == c1) ? w1 : 0.0f;

        acc0 = __builtin_amdgcn_wmma_f32_16x16x4_f32(
            false, a0, false, b, (short)0, acc0, false, false);
        acc1 = __builtin_amdgcn_wmma_f32_16x16x4_f32(
            false, a1, false, b, (short)0, acc1, false, false);

        lptr += 4;
        fptr += 4 * (long)FEAT_CH;
    }

    // D layout: VGPR r -> class (r + 8*hi), col = l16
    float* base = sums + colb + l16;
#pragma unroll
    for (int r = 0; r < 8; ++r) {
        const int m = r + 8 * hi;
        atomicAdd(base + (long)m * FEAT_CH,        acc0[r]);
        atomicAdd(base + (long)(16 + m) * FEAT_CH, acc1[r]);
    }
}

// --------------------------------------------- scalar tail (non-divisible sizes)
__global__ void tail_kernel(const float* __restrict__ feat,
                            const int*   __restrict__ labels,
                            const int*   __restrict__ ignore_p,
                            int start, int n,
                            float* __restrict__ sums) {
    const int i = start + blockIdx.x;
    if (i >= n) return;
    const int lab = labels[i];
    if (lab == *ignore_p || lab < 0 || lab >= NUM_CLASSES) return;
    const float* row = feat + (long)i * FEAT_CH;
    for (int a = threadIdx.x; a < FEAT_CH; a += blockDim.x)
        atomicAdd(&sums[(long)lab * FEAT_CH + a], row[a]);
}

// ---------------------------------------------------------------- finalize out
__global__ void finalize_kernel(const float* __restrict__ sums,
                                const float* __restrict__ counts,
                                float* __restrict__ out) {
    const int CA = NUM_CLASSES * FEAT_CH;
    const int i = blockIdx.x * blockDim.x + threadIdx.x;
    if (i < CA) {
        const int c = i / FEAT_CH;
        const int a = i - c * FEAT_CH;
        const float cnt = counts[c];
        const float amount = (cnt == 0.0f) ? 1.0f : cnt;
        out[i]      = sums[(long)c * FEAT_CH + a] / amount;  // mean
        out[CA + i] = cnt;                                   // sum_weight
    }
    if (i < NUM_CLASSES) out[2 * CA + i] = counts[i];        // class_dist
}

extern "C" void kernel_launch(void* const* d_in, const int* in_sizes, int n_in,
                              void* d_out, int out_size, void* d_ws, size_t ws_size,
                              hipStream_t stream) {
    const float* feat     = (const float*)d_in[0];
    const int*   labels   = (const int*)  d_in[1];
    const int*   ignore_p = (const int*)  d_in[2];
    const int n = in_sizes[1];                 // N pixels

    float* sums   = (float*)d_ws;              // C_PAD x FEAT_CH
    float* counts = sums + C_PAD * FEAT_CH;    // C_PAD
    const int nz = C_PAD * FEAT_CH + C_PAD;

    zero_ws_kernel<<<(nz + 255) / 256, 256, 0, stream>>>((float*)d_ws, nz);

    count_kernel<<<256, 256, 0, stream>>>(labels, n, ignore_p, counts);

    const int full_tasks = n / CHUNK;
    const int gx = full_tasks / WPB;
    const int covered = gx * WPB * CHUNK;
    if (gx > 0) {
        dim3 grid(gx, COL_TILES, 1);
        segsum_wmma_kernel<<<grid, 256, 0, stream>>>(feat, labels, ignore_p, sums);
    }
    if (covered < n) {
        tail_kernel<<<n - covered, 256, 0, stream>>>(feat, labels, ignore_p,
                                                     covered, n, sums);
    }

    const int CA = NUM_CLASSES * FEAT_CH;
    finalize_kernel<<<(CA + 255) / 256, 256, 0, stream>>>(sums, counts,
                                                          (float*)d_out);
}